// CustomMaxPool2d_20942260535372
// MI455X (gfx1250) — compile-verified
//
#include <hip/hip_runtime.h>

// 2x2 stride-2 max pool, NCHW fp32, H=W=224 -> 112.
// Memory-bound streamer: one thread produces one float2 of output from two
// aligned float4 non-temporal loads (row 2h and row 2h+1). Output store is
// purely linear (offset = 2*t).

typedef float v4f __attribute__((ext_vector_type(4)));
typedef float v2f __attribute__((ext_vector_type(2)));

#define W_IN    224u
#define PLANE   (224u * 224u)   // 50176
#define W_OUT   112u
#define CHUNKS_PER_ROW 56u      // 112 outputs / 2 per chunk

__global__ __launch_bounds__(256) void maxpool2x2_nt_kernel(
    const float* __restrict__ in, float* __restrict__ out, unsigned total_chunks)
{
    unsigned t = blockIdx.x * 256u + threadIdx.x;
    if (t >= total_chunks) return;

    // chunk -> (plane, h_out, within-row chunk); output offset is simply 2*t.
    unsigned within = t % CHUNKS_PER_ROW;      // const-div -> mulhi+shift
    unsigned row    = t / CHUNKS_PER_ROW;      // row = plane*112 + h_out
    unsigned h      = row % 112u;
    unsigned plane  = row / 112u;

    unsigned in_off = plane * PLANE + (2u * h) * W_IN + within * 4u;

    // Two aligned 16B loads, marked non-temporal (stream > L2 capacity,
    // each byte read exactly once) -> global_load_b128 th:NT.
    v4f a = __builtin_nontemporal_load((const v4f*)(in + in_off));
    v4f b = __builtin_nontemporal_load((const v4f*)(in + in_off + W_IN));

    v2f r;
    r.x = fmaxf(fmaxf(a.x, a.y), fmaxf(b.x, b.y));
    r.y = fmaxf(fmaxf(a.z, a.w), fmaxf(b.z, b.w));

    // Dense linear store, non-temporal (write-once, never re-read).
    __builtin_nontemporal_store(r, (v2f*)(out + 2u * t));
}

extern "C" void kernel_launch(void* const* d_in, const int* in_sizes, int n_in,
                              void* d_out, int out_size, void* d_ws, size_t ws_size,
                              hipStream_t stream) {
    const float* x = (const float*)d_in[0];
    float* y = (float*)d_out;

    // planes = N*C derived from the actual input size; H=W=224 fixed by reference.
    unsigned planes = (unsigned)((unsigned long long)in_sizes[0] / PLANE);   // 3072
    unsigned total_chunks = planes * 112u * CHUNKS_PER_ROW;                  // 19,267,584

    dim3 block(256);
    dim3 grid((total_chunks + 255u) / 256u);
    maxpool2x2_nt_kernel<<<grid, block, 0, stream>>>(x, y, total_chunks);
}